// PMF_32796370272269
// MI455X (gfx1250) — compile-verified
//
#include <hip/hip_runtime.h>

// PMF gather-dot:
//   inputs : U [1e6 x 5] f32, V [1e6 x 5] f32, u_index [16e6] i64, v_index [16e6] i64
//   outputs: uv [N x 4] f32, preds [N] f32 (concatenated flat in d_out)
//
// Memory-bound streaming gather. U/V tables (40 MB) live in the 192 MB L2;
// index reads and all output writes use non-temporal hints so the 320 MB
// output stream does not evict the tables. global_prefetch_b8 runs ahead on
// the index streams.

#define LATENT 5

// Native clang vector type (HIP's float4 is a class and is rejected by
// __builtin_nontemporal_store). Same 16-byte layout.
typedef float v4f __attribute__((ext_vector_type(4)));

__global__ __launch_bounds__(256) void pmf_gather_dot_kernel(
    const float* __restrict__ U,
    const float* __restrict__ V,
    const long long* __restrict__ u_idx,
    const long long* __restrict__ v_idx,
    float* __restrict__ uv_out,      // [n*4]
    float* __restrict__ preds_out,   // [n]
    int n)
{
    const int i = blockIdx.x * 256 + threadIdx.x;
    if (i >= n) return;

    // Prefetch the index streams well ahead of the wavefront
    // (emits global_prefetch_b8; speculative, silently dropped if OOB,
    //  but guard anyway to keep addresses sane).
    const int pf = i + 32768;
    if (pf < n) {
        __builtin_prefetch(&u_idx[pf], 0, 1);
        __builtin_prefetch(&v_idx[pf], 0, 1);
    }

    // One-pass streamed index loads: non-temporal so they don't pollute L2
    // (L2 must stay reserved for the U/V gather tables).
    const long long u = __builtin_nontemporal_load(&u_idx[i]);
    const long long v = __builtin_nontemporal_load(&v_idx[i]);

    // Rows are 20 B apart -> only 4 B alignment guaranteed: scalar b32 gathers.
    // These are the L2-resident accesses; keep default (RT) caching.
    const float* __restrict__ ur = U + (size_t)(unsigned long long)u * LATENT;
    const float* __restrict__ vr = V + (size_t)(unsigned long long)v * LATENT;

    const float u0 = ur[0], u1 = ur[1], u2 = ur[2], u3 = ur[3];
    const float v0 = vr[0], v1 = vr[1], v2 = vr[2], v3 = vr[3];

    v4f uv;
    uv.x = u0 * v0;
    uv.y = u1 * v1;
    uv.z = u2 * v2;
    uv.w = u3 * v3;

    // Pure streaming output (320 MB/call): non-temporal b128 + b32 stores,
    // perfectly coalesced (lane l writes bytes [16l, 16l+16) of the wave's
    // 512 B contiguous chunk).
    __builtin_nontemporal_store(uv, (v4f*)uv_out + i);

    const float pred = (uv.x + uv.y) + (uv.z + uv.w);
    __builtin_nontemporal_store(pred, preds_out + i);
}

extern "C" void kernel_launch(void* const* d_in, const int* in_sizes, int n_in,
                              void* d_out, int out_size, void* d_ws, size_t ws_size,
                              hipStream_t stream)
{
    const float*     U  = (const float*)d_in[0];
    const float*     V  = (const float*)d_in[1];
    const long long* ui = (const long long*)d_in[2];  // int64 indices
    const long long* vi = (const long long*)d_in[3];  // int64 indices

    const int n = in_sizes[2];                 // number of pairs (16M)

    float* uv_out    = (float*)d_out;          // first n*4 floats
    float* preds_out = uv_out + (size_t)n * 4; // next n floats

    const int block = 256;                     // 8 wave32 waves per workgroup
    const int grid  = (n + block - 1) / block;

    pmf_gather_dot_kernel<<<grid, block, 0, stream>>>(
        U, V, ui, vi, uv_out, preds_out, n);
}